// MultiHeadAttentionQuantum_65481071397526
// MI455X (gfx1250) — compile-verified
//
#include <hip/hip_runtime.h>
#include <hip/hip_bf16.h>
#include <math.h>

// ---------------------------------------------------------------------------
// Quantum multi-head attention, analytically collapsed:
//   z[h][w] = prod cos(x+theta) prefix (CNOT-ring parity telescopes),
//   per-token 16x16 head attention, scrambled reshape, then
//   [4096x128] @ w_combine^T via V_WMMA_F32_16X16X4_F32 (fp32 WMMA).
// ---------------------------------------------------------------------------

typedef __attribute__((ext_vector_type(2))) float v2f;
typedef __attribute__((ext_vector_type(8))) float v8f;

#define N_TOK   4096   // B*S = 8*512
#define S_LEN   512
#define E_DIM   128
#define N_HEAD  16
#define D_K     8

// ---------------------------------------------------------------------------
// Kernel A: one block (128 threads) per token.
//   thread t -> head h = t>>3, wire w = t&7  (t == e index into x[b,s,:])
// ---------------------------------------------------------------------------
__global__ __launch_bounds__(128) void qlayer_attn_kernel(
    const float* __restrict__ x,      // [B,S,E] flattened
    const float* __restrict__ theta,  // [8]
    float* __restrict__ pre)          // [B, 512, 128] scrambled pre-combine
{
    const int token = blockIdx.x;          // b*512 + s
    const int b = token >> 9;
    const int s = token & 511;
    const int t = threadIdx.x;             // 0..127
    const int h = t >> 3;
    const int d = t & 7;

    __shared__ float cth[N_HEAD][D_K];     // cos(x+theta)
    __shared__ float z[N_HEAD][D_K];       // <Z_w> per head
    __shared__ float att[N_HEAD][N_HEAD];  // scores -> softmax weights

    // --- cosines ---
    const float* xr = x + token * E_DIM;
    cth[h][d] = __cosf(xr[t] + theta[d]);
    __syncthreads();

    // --- prefix products (CNOT ring parities): 16 threads, one per head ---
    if (t < N_HEAD) {
        float zz[D_K];
        float p = cth[t][0];
        #pragma unroll
        for (int w = 1; w < D_K; ++w) { p *= cth[t][w]; zz[w] = p; }
        float q = 1.0f;
        #pragma unroll
        for (int j = 1; j < D_K; ++j) q *= cth[t][j];
        zz[0] = q;
        #pragma unroll
        for (int w = 0; w < D_K; ++w) z[t][w] = zz[w];
    }
    __syncthreads();

    // --- scores[h][u] = z[h].z[u] / sqrt(8): 256 entries, 2 per thread ---
    const float inv_sqrt_dk = 0.35355339059327373f; // 1/sqrt(8)
    #pragma unroll
    for (int r = 0; r < 2; ++r) {
        int idx = 2 * t + r;               // 0..255
        int hh = idx >> 4, uu = idx & 15;
        float acc = 0.0f;
        #pragma unroll
        for (int k = 0; k < D_K; ++k) acc += z[hh][k] * z[uu][k];
        att[hh][uu] = acc * inv_sqrt_dk;
    }
    __syncthreads();

    // --- softmax over u: 16 threads, one row each ---
    if (t < N_HEAD) {
        float m = att[t][0];
        #pragma unroll
        for (int u = 1; u < N_HEAD; ++u) m = fmaxf(m, att[t][u]);
        float sum = 0.0f;
        float e[N_HEAD];
        #pragma unroll
        for (int u = 0; u < N_HEAD; ++u) { e[u] = __expf(att[t][u] - m); sum += e[u]; }
        float inv = 1.0f / sum;
        #pragma unroll
        for (int u = 0; u < N_HEAD; ++u) att[t][u] = e[u] * inv;
    }
    __syncthreads();

    // --- out[h][d] = sum_u att[h][u] * z[u][d], one element per thread ---
    float o = 0.0f;
    #pragma unroll
    for (int u = 0; u < N_HEAD; ++u) o += att[h][u] * z[u][d];

    // --- source-faithful scrambled reshape [B,H,S,8] -> [B,512,128] ---
    //   i = h*32 + s/16 ; e = (s%16)*8 + d
    const int i = h * 32 + (s >> 4);
    const int e = ((s & 15) << 3) + d;
    pre[(b * S_LEN + i) * E_DIM + e] = o;
}

// ---------------------------------------------------------------------------
// Kernel B: out[m][n] = sum_k pre[m][k] * w_combine[n][k]
//   M = 4096, N = 128, K = 128. 8 waves/block; block = one 16-row M tile
//   (staged in LDS), wave = one 16-col N tile, K loop = 32 x wmma 16x16x4.
// Fragment layouts per CDNA5 ISA 7.12.2 (32-bit A 16x4; C/D 16x16).
// ---------------------------------------------------------------------------
__global__ __launch_bounds__(256) void combine_gemm_wmma(
    const float* __restrict__ pre,   // [4096,128]
    const float* __restrict__ wc,    // w_combine [128,128], row-major
    float* __restrict__ out)         // [4096,128]
{
    __shared__ float As[16 * E_DIM]; // one M tile, 8 KB

    const int mtile = blockIdx.x;            // 0..255
    // cooperative stage of A tile (16 x 128)
    for (int idx = threadIdx.x; idx < 16 * E_DIM; idx += 256) {
        int r = idx >> 7, col = idx & 127;
        As[idx] = pre[(mtile * 16 + r) * E_DIM + col];
    }
    __syncthreads();

    const int wave = threadIdx.x >> 5;       // N tile 0..7
    const int lane = threadIdx.x & 31;
    const int lm    = lane & 15;             // row (A) / col (B) within tile
    const int khalf = (lane >> 4) << 1;      // 0 or 2: which K pair this half-wave holds

    v8f acc = {0.f, 0.f, 0.f, 0.f, 0.f, 0.f, 0.f, 0.f};

    // B[k][n] = w_combine[n][k]; lane holds {B[k0][n], B[k0+1][n]} = contiguous
    // pair within one w_combine row -> coalesced, L2-resident (64 KB matrix).
    const float* wrow = wc + (wave * 16 + lm) * E_DIM;

    #pragma unroll 8
    for (int kk = 0; kk < 32; ++kk) {
        const int kg = (kk << 2) + khalf;
        v2f a, bfrag;
        a.x = As[lm * E_DIM + kg];
        a.y = As[lm * E_DIM + kg + 1];
        bfrag.x = wrow[kg];
        bfrag.y = wrow[kg + 1];
        acc = __builtin_amdgcn_wmma_f32_16x16x4_f32(
            /*neg_a=*/false, a, /*neg_b=*/false, bfrag,
            /*c_mod=*/(short)0, acc, /*reuse_a=*/false, /*reuse_b=*/false);
    }

    // D layout: VGPR i -> M = i (lanes 0-15) / i+8 (lanes 16-31), N = lane%16
    const int n = wave * 16 + lm;
    const int mbase = (lane >> 4) << 3;
    #pragma unroll
    for (int i = 0; i < 8; ++i) {
        const int m = mtile * 16 + mbase + i;
        out[m * E_DIM + n] = acc[i];
    }
}

// ---------------------------------------------------------------------------
extern "C" void kernel_launch(void* const* d_in, const int* in_sizes, int n_in,
                              void* d_out, int out_size, void* d_ws, size_t ws_size,
                              hipStream_t stream) {
    const float* x     = (const float*)d_in[0];  // [8,512,128]
    const float* theta = (const float*)d_in[1];  // [8]
    const float* wc    = (const float*)d_in[2];  // [128,128]
    float* out = (float*)d_out;                  // [8,512,128]
    float* pre = (float*)d_ws;                   // 4096*128 floats = 2 MB scratch

    qlayer_attn_kernel<<<N_TOK, 128, 0, stream>>>(x, theta, pre);
    combine_gemm_wmma<<<N_TOK / 16, 256, 0, stream>>>(pre, wc, out);
}